// VGNAEencoder_26465588478695
// MI455X (gfx1250) — compile-verified
//
#include <hip/hip_runtime.h>
#include <hip/hip_bf16.h>

// VGNAE encoder forward for MI455X (gfx1250, wave32).
// Bandwidth-bound SpMMs + two small dense GEMMs done in fp32 on the WMMA pipe
// (V_WMMA_F32_16X16X4_F32) to match reference precision.

#define D_IN   128
#define D_HID  128
#define D_OUT  64
#define BN_EPS 1e-5f
#define LRELU_SLOPE 0.01f
#define NORM_SCALE 1.8f

typedef __attribute__((ext_vector_type(2))) float v2f;
typedef __attribute__((ext_vector_type(8))) float v8f;

// ---------------------------------------------------------------- init: deg=1, stats=0
__global__ void vg_init_kernel(float* __restrict__ deg, float* __restrict__ stats, int n) {
  int i = blockIdx.x * blockDim.x + threadIdx.x;
  if (i < n) deg[i] = 1.0f;
  if (i < 256) stats[i] = 0.0f;
}

// ---------------------------------------------------------------- deg[dst] += 1
__global__ void vg_deg_kernel(const int* __restrict__ dst, float* __restrict__ deg, int e) {
  int i = blockIdx.x * blockDim.x + threadIdx.x;
  if (i < e) atomicAdd(&deg[dst[i]], 1.0f);
}

// ---------------------------------------------------------------- dinv = rsqrt(deg), in place
__global__ void vg_dinv_kernel(float* __restrict__ deg, int n) {
  int i = blockIdx.x * blockDim.x + threadIdx.x;
  if (i < n) deg[i] = rsqrtf(deg[i]);
}

// ---------------------------------------------------------------- C[M,Ncols] = A[M,K] @ B[K,Ncols] + bias
// One wave computes one 16x16 output tile via V_WMMA_F32_16X16X4_F32.
// A layout (16x4 f32): lane<16 -> M=lane, v0=K0,v1=K1 ; lane>=16 -> v0=K2,v1=K3.
// B layout (4x16 f32): mirrors A over N: lane<16 -> N=lane, v0=K0,v1=K1 ; hi half K2,K3.
// C/D layout: VGPR j -> M = (lane>=16 ? 8 : 0)+j, N = lane&15.
__global__ void vg_gemm_wmma_kernel(const float* __restrict__ A, const float* __restrict__ B,
                                    const float* __restrict__ bias, float* __restrict__ C,
                                    int M, int K, int Ncols) {
  const int wavesPerBlock = blockDim.x >> 5;
  int w = blockIdx.x * wavesPerBlock + (threadIdx.x >> 5);
  int tilesN = Ncols >> 4;
  int totalTiles = (M >> 4) * tilesN;
  if (w >= totalTiles) return;          // wave-uniform: EXEC stays all-1s for WMMA
  int tm = w / tilesN;
  int tn = w - tm * tilesN;
  int lane = threadIdx.x & 31;
  int m  = lane & 15;
  int hi = lane >> 4;

  v8f acc = {};
  const float* arow = A + (size_t)(tm * 16 + m) * (size_t)K;
  const float* bcol = B + tn * 16 + m;
  for (int k0 = 0; k0 < K; k0 += 4) {
    int kb = k0 + 2 * hi;
    v2f a = *(const v2f*)(arow + kb);           // 8B-aligned pair load
    v2f b;
    b.x = bcol[(size_t)kb * (size_t)Ncols];
    b.y = bcol[(size_t)(kb + 1) * (size_t)Ncols];
    acc = __builtin_amdgcn_wmma_f32_16x16x4_f32(false, a, false, b,
                                                (short)0, acc, false, false);
  }
  float bv = bias[tn * 16 + m];
  int rbase = tm * 16 + hi * 8;
#pragma unroll
  for (int j = 0; j < 8; ++j) {
    C[(size_t)(rbase + j) * (size_t)Ncols + (size_t)(tn * 16 + m)] = acc[j] + bv;
  }
}

// ---------------------------------------------------------------- out = h * dinv^2 (self-loop), D=128
__global__ void vg_selfloop128_kernel(const float* __restrict__ h, const float* __restrict__ dinv,
                                      float* __restrict__ out, int total) {
  int i = blockIdx.x * blockDim.x + threadIdx.x;
  if (i >= total) return;
  float di = dinv[i >> 7];
  out[i] = h[i] * di * di;
}

// ---------------------------------------------------------------- scatter: out[dst] += w * in[src], D=128
// 32 lanes per edge, float4 per lane, global f32 atomics.
__global__ void vg_spmm128_kernel(const float* __restrict__ in, const int* __restrict__ src,
                                  const int* __restrict__ dst, const float* __restrict__ dinv,
                                  float* __restrict__ out, int E) {
  int tid = blockIdx.x * blockDim.x + threadIdx.x;
  int e = tid >> 5;
  if (e >= E) return;
  int lane = tid & 31;
  int s = src[e], d = dst[e];
  float w = dinv[s] * dinv[d];
  float4 v = ((const float4*)(in + (size_t)s * D_HID))[lane];
  float* op = out + (size_t)d * D_HID + lane * 4;
  atomicAdd(op + 0, w * v.x);
  atomicAdd(op + 1, w * v.y);
  atomicAdd(op + 2, w * v.z);
  atomicAdd(op + 3, w * v.w);
}

// ---------------------------------------------------------------- per-channel sum / sumsq
__global__ void vg_bn_reduce_kernel(const float* __restrict__ h, float* __restrict__ stats, int n) {
  __shared__ float sh[256];
  int c    = threadIdx.x & 127;
  int half = threadIdx.x >> 7;   // 0 or 1
  float sum = 0.f, sq = 0.f;
  for (int r = blockIdx.x * 2 + half; r < n; r += gridDim.x * 2) {
    float v = h[(size_t)r * D_HID + c];
    sum += v; sq += v * v;
  }
  sh[threadIdx.x] = sum;
  __syncthreads();
  if (half == 0) atomicAdd(&stats[c], sh[c] + sh[c + 128]);
  __syncthreads();
  sh[threadIdx.x] = sq;
  __syncthreads();
  if (half == 0) atomicAdd(&stats[128 + c], sh[c] + sh[c + 128]);
}

// ---------------------------------------------------------------- stats -> per-channel scale/shift
__global__ void vg_bn_finalize_kernel(float* __restrict__ stats, const float* __restrict__ gamma,
                                      const float* __restrict__ beta, int n) {
  int c = threadIdx.x;
  if (c < 128) {
    float invn = 1.0f / (float)n;
    float mean = stats[c] * invn;
    float var  = stats[128 + c] * invn - mean * mean;
    float a = gamma[c] * rsqrtf(var + BN_EPS);
    stats[c]       = a;                    // scale
    stats[128 + c] = beta[c] - mean * a;   // shift
  }
}

// ---------------------------------------------------------------- BN + LeakyReLU, elementwise
__global__ void vg_bn_lrelu_kernel(const float* __restrict__ hagg, const float* __restrict__ stats,
                                   float* __restrict__ out, int total) {
  int i = blockIdx.x * blockDim.x + threadIdx.x;
  if (i >= total) return;
  int c = i & 127;
  float v = stats[c] * hagg[i] + stats[128 + c];
  out[i] = (v > 0.f) ? v : LRELU_SLOPE * v;
}

// ---------------------------------------------------------------- rns[i] = 1.8 / max(||l_i||, 1e-12), wave per row
__global__ void vg_rownorm_kernel(const float* __restrict__ l, float* __restrict__ rns, int n) {
  int row = blockIdx.x * (blockDim.x >> 5) + (threadIdx.x >> 5);
  if (row >= n) return;
  int lane = threadIdx.x & 31;
  const float* rp = l + (size_t)row * D_OUT;
  float a0 = rp[lane], a1 = rp[lane + 32];
  float s = a0 * a0 + a1 * a1;
#pragma unroll
  for (int m = 16; m > 0; m >>= 1) s += __shfl_xor(s, m, 32);
  if (lane == 0) rns[row] = NORM_SCALE / fmaxf(sqrtf(s), 1e-12f);
}

// ---------------------------------------------------------------- self-loop init for mu and logstd, D=64
__global__ void vg_selfloop64_kernel(const float* __restrict__ l, const float* __restrict__ dinv,
                                     const float* __restrict__ rns, float* __restrict__ mu,
                                     float* __restrict__ logstd, int total) {
  int i = blockIdx.x * blockDim.x + threadIdx.x;
  if (i >= total) return;
  int row = i >> 6;
  float di = dinv[row];
  float d2 = di * di;
  float v = l[i];
  mu[i]     = v * d2;
  logstd[i] = v * rns[row] * d2;
}

// ---------------------------------------------------------------- fused mu/logstd scatter, D=64
// 16 lanes per edge, float4 per lane; l[src] gathered once, used for both outputs.
__global__ void vg_spmm64_kernel(const float* __restrict__ l, const int* __restrict__ src,
                                 const int* __restrict__ dst, const float* __restrict__ dinv,
                                 const float* __restrict__ rns, float* __restrict__ mu,
                                 float* __restrict__ logstd, int E) {
  int tid = blockIdx.x * blockDim.x + threadIdx.x;
  int e = tid >> 4;
  if (e >= E) return;
  int q = tid & 15;
  int s = src[e], d = dst[e];
  float w  = dinv[s] * dinv[d];
  float sc = w * rns[s];
  float4 v = ((const float4*)(l + (size_t)s * D_OUT))[q];
  float* mp = mu     + (size_t)d * D_OUT + q * 4;
  float* lp = logstd + (size_t)d * D_OUT + q * 4;
  atomicAdd(mp + 0, w * v.x);  atomicAdd(mp + 1, w * v.y);
  atomicAdd(mp + 2, w * v.z);  atomicAdd(mp + 3, w * v.w);
  atomicAdd(lp + 0, sc * v.x); atomicAdd(lp + 1, sc * v.y);
  atomicAdd(lp + 2, sc * v.z); atomicAdd(lp + 3, sc * v.w);
}

// ---------------------------------------------------------------- zeta = mu + noise * exp(logstd)
__global__ void vg_zeta_kernel(const float* __restrict__ mu, const float* __restrict__ logstd,
                               const float* __restrict__ noise, float* __restrict__ zeta, int total) {
  int i = blockIdx.x * blockDim.x + threadIdx.x;
  if (i >= total) return;
  zeta[i] = mu[i] + noise[i] * expf(logstd[i]);
}

extern "C" void kernel_launch(void* const* d_in, const int* in_sizes, int n_in,
                              void* d_out, int out_size, void* d_ws, size_t ws_size,
                              hipStream_t stream) {
  const float* x     = (const float*)d_in[0];
  const int*   esrc  = (const int*)  d_in[1];
  const int*   edst  = (const int*)  d_in[2];
  const float* W1    = (const float*)d_in[3];
  const float* b1    = (const float*)d_in[4];
  const float* gamma = (const float*)d_in[5];
  const float* beta  = (const float*)d_in[6];
  const float* Wmu   = (const float*)d_in[7];
  const float* bmu   = (const float*)d_in[8];
  const float* noise = (const float*)d_in[9];

  const int N = in_sizes[0] / D_IN;   // 100000
  const int E = in_sizes[1];          // 1600000

  // Workspace layout (floats), ~129 MB total
  float* ws = (float*)d_ws;
  size_t npad = ((size_t)N + 255) & ~(size_t)255;
  size_t off = 0;
  float* dinv  = ws + off; off += npad;                 // deg -> dinv in place
  float* stats = ws + off; off += 256;                  // sum/sumsq -> scale/shift
  float* rns   = ws + off; off += npad;                 // 1.8 / row-norm of l
  float* h     = ws + off; off += (size_t)N * D_HID;    // x@W1+b1, later LReLU(BN(hagg)) output
  float* hagg  = ws + off; off += (size_t)N * D_HID;    // appnp(h)
  float* lbuf  = ws + off; off += (size_t)N * D_OUT;    // h@Wmu+bmu

  float* mu_out   = (float*)d_out;
  float* ls_out   = mu_out + (size_t)N * D_OUT;
  float* zeta_out = ls_out + (size_t)N * D_OUT;

  const int B = 256;
  auto cdiv = [](int a, int b) { return (a + b - 1) / b; };

  // 1) degree / dinv
  vg_init_kernel<<<cdiv(N, B), B, 0, stream>>>(dinv, stats, N);
  vg_deg_kernel<<<cdiv(E, B), B, 0, stream>>>(edst, dinv, E);
  vg_dinv_kernel<<<cdiv(N, B), B, 0, stream>>>(dinv, N);

  // 2) h = x @ W1 + b1   (WMMA f32, one 16x16 tile per wave, 8 waves/block)
  {
    int tiles = (N / 16) * (D_HID / 16);
    vg_gemm_wmma_kernel<<<cdiv(tiles, B / 32), B, 0, stream>>>(x, W1, b1, h, N, D_IN, D_HID);
  }

  // 3) hagg = appnp(h)
  vg_selfloop128_kernel<<<cdiv(N * D_HID, B), B, 0, stream>>>(h, dinv, hagg, N * D_HID);
  vg_spmm128_kernel<<<cdiv(E * 32, B), B, 0, stream>>>(h, esrc, edst, dinv, hagg, E);

  // 4) BatchNorm (batch stats) + LeakyReLU -> back into h
  vg_bn_reduce_kernel<<<512, B, 0, stream>>>(hagg, stats, N);
  vg_bn_finalize_kernel<<<1, 128, 0, stream>>>(stats, gamma, beta, N);
  vg_bn_lrelu_kernel<<<cdiv(N * D_HID, B), B, 0, stream>>>(hagg, stats, h, N * D_HID);

  // 5) l = h @ Wmu + bmu  (WMMA f32)
  {
    int tiles = (N / 16) * (D_OUT / 16);
    vg_gemm_wmma_kernel<<<cdiv(tiles, B / 32), B, 0, stream>>>(h, Wmu, bmu, lbuf, N, D_HID, D_OUT);
  }

  // 6) row norms for logstd branch (stores 1.8/||l_i|| only; avoids a second N x 64 buffer)
  vg_rownorm_kernel<<<cdiv(N, B / 32), B, 0, stream>>>(lbuf, rns, N);

  // 7) mu = appnp(l), logstd = appnp(l * rns[row]) — fused self-loop + fused scatter
  vg_selfloop64_kernel<<<cdiv(N * D_OUT, B), B, 0, stream>>>(lbuf, dinv, rns, mu_out, ls_out, N * D_OUT);
  vg_spmm64_kernel<<<cdiv(E * 16, B), B, 0, stream>>>(lbuf, esrc, edst, dinv, rns, mu_out, ls_out, E);

  // 8) zeta = mu + noise * exp(logstd)
  vg_zeta_kernel<<<cdiv(N * D_OUT, B), B, 0, stream>>>(mu_out, ls_out, noise, zeta_out, N * D_OUT);
}